// ResidualBlock_65962107732684
// MI455X (gfx1250) — compile-verified
//
#include <hip/hip_runtime.h>
#include <math.h>

// ---------------- problem constants (match reference) ----------------
constexpr int HIDDEN  = 1024;
constexpr int D_INNER = 2048;
constexpr int STATE   = 16;
constexpr int DT_RANK = 64;
constexpr int D_CONV  = 4;
constexpr int BATCH   = 2;
constexpr int SEQ     = 2048;
constexpr int ROWS    = BATCH * SEQ;          // 4096
constexpr int XDBL    = DT_RANK + 2 * STATE;  // 96
constexpr float EPS   = 1e-5f;

typedef float v2f __attribute__((ext_vector_type(2)));
typedef float v8f __attribute__((ext_vector_type(8)));
typedef int   v4i __attribute__((ext_vector_type(4)));

// ---- async global->LDS (gfx1250 GLOBAL_LOAD_ASYNC_TO_LDS_B128) ----
#if defined(__has_builtin)
# if __has_builtin(__builtin_amdgcn_global_load_async_to_lds_b128)
#  define HAS_ASYNC_LDS 1
# endif
#endif
#ifndef HAS_ASYNC_LDS
# define HAS_ASYNC_LDS 0
#endif

// param types per hipcc diagnostic: (AS1 int4*, AS3 int4*, imm int, imm int)
#define GAS1(p) ((__attribute__((address_space(1))) v4i*)(p))
#define LAS3(p) ((__attribute__((address_space(3))) v4i*)(p))

__device__ __forceinline__ void wait_async0() {
#if defined(__has_builtin) && __has_builtin(__builtin_amdgcn_s_wait_asynccnt)
  __builtin_amdgcn_s_wait_asynccnt(0);
#else
  asm volatile("s_wait_asynccnt 0x0" ::: "memory");
#endif
}

// ---------------- RMSNorm: one block per row ----------------
__global__ __launch_bounds__(256)
void rmsnorm_kernel(const float* __restrict__ x, const float* __restrict__ w,
                    float* __restrict__ out) {
  const int row = blockIdx.x;
  const int tid = threadIdx.x;
  const float* xr = x + (size_t)row * HIDDEN;
  float ss = 0.f;
  #pragma unroll
  for (int i = tid; i < HIDDEN; i += 256) { float v = xr[i]; ss = fmaf(v, v, ss); }
  #pragma unroll
  for (int off = 16; off > 0; off >>= 1) ss += __shfl_xor(ss, off, 32);
  __shared__ float red[8];
  if ((tid & 31) == 0) red[tid >> 5] = ss;
  __syncthreads();
  float total = 0.f;
  #pragma unroll
  for (int i = 0; i < 8; ++i) total += red[i];
  const float scale = rsqrtf(total * (1.0f / HIDDEN) + EPS);
  #pragma unroll
  for (int i = tid; i < HIDDEN; i += 256)
    out[(size_t)row * HIDDEN + i] = xr[i] * scale * w[i];
}

// ---------------- LDS double-buffered fp32 WMMA GEMM ----------------
// C[M,N] = A[M,K(lda)] * W[N,K]^T ; block tile 128(M) x 64(N), K-chunk 32
// mode 0: store   mode 1: softplus(val + bias[n])   mode 2: val + addend
constexpr int KC    = 32;
constexpr int PAD   = 36;           // bank-conflict-free & 16B-aligned rows
constexpr int A_SZ  = 128 * PAD;    // 4608 floats
constexpr int B_SZ  = 64  * PAD;    // 2304 floats
constexpr int BUFSZ = A_SZ + B_SZ;  // 6912 floats per buffer

__device__ __forceinline__
void gemm_stage_issue(const float* __restrict__ A, int lda, int mBase,
                      const float* __restrict__ W, int K, int N, int n0,
                      int k0, float* dst, int tid
#if !HAS_ASYNC_LDS
                      , float4* ar, float4* br
#endif
                      ) {
  // A tile: 128 rows x 32 cols = 1024 float4 slots, 4 per thread
  #pragma unroll
  for (int i = 0; i < 4; ++i) {
    const int s = tid + i * 256, row = s >> 3, c4 = s & 7;
    const float* g = A + (size_t)(mBase + row) * lda + k0 + c4 * 4;
#if HAS_ASYNC_LDS
    __builtin_amdgcn_global_load_async_to_lds_b128(
        GAS1(g), LAS3(dst + row * PAD + c4 * 4), 0, 0);
#else
    ar[i] = *(const float4*)g;
#endif
  }
  // B tile: 64 rows x 32 cols = 512 float4 slots, 2 per thread
  #pragma unroll
  for (int i = 0; i < 2; ++i) {
    const int s = tid + i * 256, row = s >> 3, c4 = s & 7;
    int wr = n0 + row; wr = (wr < N) ? wr : (N - 1);   // clamp (guarded store later)
    const float* g = W + (size_t)wr * K + k0 + c4 * 4;
#if HAS_ASYNC_LDS
    __builtin_amdgcn_global_load_async_to_lds_b128(
        GAS1(g), LAS3(dst + A_SZ + row * PAD + c4 * 4), 0, 0);
#else
    br[i] = *(const float4*)g;
#endif
  }
}

__device__ __forceinline__
void gemm_stage_finish(float* dst, int tid
#if !HAS_ASYNC_LDS
                       , const float4* ar, const float4* br
#endif
                       ) {
#if HAS_ASYNC_LDS
  wait_async0();
#else
  #pragma unroll
  for (int i = 0; i < 4; ++i) {
    const int s = tid + i * 256, row = s >> 3, c4 = s & 7;
    *(float4*)(dst + row * PAD + c4 * 4) = ar[i];
  }
  #pragma unroll
  for (int i = 0; i < 2; ++i) {
    const int s = tid + i * 256, row = s >> 3, c4 = s & 7;
    *(float4*)(dst + A_SZ + row * PAD + c4 * 4) = br[i];
  }
#endif
}

__global__ __launch_bounds__(256)
void gemm_wmma_f32(const float* __restrict__ A, int lda,
                   const float* __restrict__ W,   // [N,K] row-major
                   float* __restrict__ C,
                   int M, int N, int K,
                   const float* __restrict__ bias,
                   const float* __restrict__ addend,
                   int mode) {
  __shared__ float smem[2 * BUFSZ];
  const int tid  = threadIdx.x;
  const int wave = tid >> 5;
  const int lane = tid & 31;
  const int half = lane >> 4;   // 0: K 0..1 / M 0..7 ; 1: K 2..3 / M 8..15
  const int l    = lane & 15;
  const int mBase = blockIdx.y * 128;          // M always a multiple of 128 here
  const int n0    = blockIdx.x * 64;
  const int nc    = K / KC;                     // K is a multiple of 32

#if !HAS_ASYNC_LDS
  float4 ar[4], br[2];
#endif

  // prologue: stage chunk 0 into buffer 0
#if HAS_ASYNC_LDS
  gemm_stage_issue(A, lda, mBase, W, K, N, n0, 0, &smem[0], tid);
  gemm_stage_finish(&smem[0], tid);
#else
  gemm_stage_issue(A, lda, mBase, W, K, N, n0, 0, &smem[0], tid, ar, br);
  gemm_stage_finish(&smem[0], tid, ar, br);
#endif
  __syncthreads();

  v8f acc[4] = {v8f{}, v8f{}, v8f{}, v8f{}};

  for (int c = 0; c < nc; ++c) {
    float* nxt = &smem[((c + 1) & 1) * BUFSZ];
    if (c + 1 < nc) {
#if HAS_ASYNC_LDS
      gemm_stage_issue(A, lda, mBase, W, K, N, n0, (c + 1) * KC, nxt, tid);
#else
      gemm_stage_issue(A, lda, mBase, W, K, N, n0, (c + 1) * KC, nxt, tid, ar, br);
#endif
    }

    const float* As = &smem[(c & 1) * BUFSZ];
    const float* Bs = As + A_SZ;
    const int arow = (wave * 16 + l) * PAD;
    #pragma unroll
    for (int kk = 0; kk < KC; kk += 4) {
      const int ko = kk + 2 * half;
      v2f a  = *(const v2f*)&As[arow + ko];
      v2f b0 = *(const v2f*)&Bs[(0 * 16 + l) * PAD + ko];
      v2f b1 = *(const v2f*)&Bs[(1 * 16 + l) * PAD + ko];
      v2f b2 = *(const v2f*)&Bs[(2 * 16 + l) * PAD + ko];
      v2f b3 = *(const v2f*)&Bs[(3 * 16 + l) * PAD + ko];
      acc[0] = __builtin_amdgcn_wmma_f32_16x16x4_f32(false, a, false, b0, (short)0, acc[0], false, false);
      acc[1] = __builtin_amdgcn_wmma_f32_16x16x4_f32(false, a, false, b1, (short)0, acc[1], false, false);
      acc[2] = __builtin_amdgcn_wmma_f32_16x16x4_f32(false, a, false, b2, (short)0, acc[2], false, false);
      acc[3] = __builtin_amdgcn_wmma_f32_16x16x4_f32(false, a, false, b3, (short)0, acc[3], false, false);
    }

    if (c + 1 < nc) {
#if HAS_ASYNC_LDS
      gemm_stage_finish(nxt, tid);
#else
      gemm_stage_finish(nxt, tid, ar, br);
#endif
    }
    __syncthreads();
  }

  // epilogue: lane l -> column n0+t*16+l ; VGPR v -> row mBase+wave*16 + v + 8*half
  const int m0 = mBase + wave * 16;
  #pragma unroll
  for (int t = 0; t < 4; ++t) {
    const int ct = n0 + t * 16 + l;
    if (ct < N) {
      #pragma unroll
      for (int v = 0; v < 8; ++v) {
        const int rowm = m0 + v + 8 * half;
        float val = acc[t][v];
        if (mode == 1) {
          val += bias[ct];
          val = (val > 20.f) ? val : log1pf(__expf(val));   // softplus
        } else if (mode == 2) {
          val += addend[(size_t)rowm * N + ct];
        }
        C[(size_t)rowm * N + ct] = val;
      }
    }
  }
}

// ---------------- causal depthwise conv1d + SiLU ----------------
__global__ __launch_bounds__(256)
void conv_silu_kernel(const float* __restrict__ xr,   // [ROWS, 2*D_INNER]
                      const float* __restrict__ cw,   // [D_INNER, D_CONV]
                      const float* __restrict__ cb,   // [D_INNER]
                      float* __restrict__ xc) {       // [ROWS, D_INNER]
  const int i = blockIdx.x * 256 + threadIdx.x;
  const int d = i & (D_INNER - 1);
  const int r = i >> 11;
  const int b = r >> 11;
  const int s = r & (SEQ - 1);
  float acc = cb[d];
  #pragma unroll
  for (int j = 0; j < D_CONV; ++j) {
    const int ss = s - (D_CONV - 1) + j;
    if (ss >= 0)
      acc = fmaf(xr[(size_t)(b * SEQ + ss) * (2 * D_INNER) + d], cw[d * D_CONV + j], acc);
  }
  xc[(size_t)i] = acc * (1.f / (1.f + __expf(-acc)));
}

// ---------------- selective scan: one lane per (b,d,n) ----------------
__global__ __launch_bounds__(256)
void selective_scan_kernel(const float* __restrict__ delta,
                           const float* __restrict__ u,
                           const float* __restrict__ xdbl,   // [ROWS,96]: B@64, C@80
                           const float* __restrict__ A_log,
                           const float* __restrict__ Dvec,
                           float* __restrict__ y) {
  const int gid = blockIdx.x * 256 + threadIdx.x;
  const int n   = gid & (STATE - 1);
  const int ch  = gid >> 4;
  const int d   = ch & (D_INNER - 1);
  const int b   = ch >> 11;
  const float a  = -__expf(A_log[d * STATE + n]);
  const float Dd = Dvec[d];
  float h = 0.f;
  const size_t rbase = (size_t)b * SEQ;
  for (int s = 0; s < SEQ; ++s) {
    const size_t r = rbase + s;
    if (s + 1 < SEQ) __builtin_prefetch(&xdbl[(r + 1) * XDBL + DT_RANK + n], 0, 1);
    const float dl = delta[r * D_INNER + d];
    const float uv = u[r * D_INNER + d];
    const float Bn = xdbl[r * XDBL + DT_RANK + n];
    const float Cn = xdbl[r * XDBL + DT_RANK + STATE + n];
    const float dA = __expf(dl * a);
    h = fmaf(dA, h, dl * Bn * uv);
    float p = h * Cn;
    p += __shfl_xor(p, 1, 32);
    p += __shfl_xor(p, 2, 32);
    p += __shfl_xor(p, 4, 32);
    p += __shfl_xor(p, 8, 32);
    if (n == 0) y[r * D_INNER + d] = fmaf(uv, Dd, p);
  }
}

// ---------------- gating: y *= silu(res) (in place) ----------------
__global__ __launch_bounds__(256)
void gate_kernel(const float* __restrict__ xr, float* __restrict__ y) {
  const int i = blockIdx.x * 256 + threadIdx.x;
  const int d = i & (D_INNER - 1);
  const int r = i >> 11;
  const float res = xr[(size_t)r * (2 * D_INNER) + D_INNER + d];
  y[(size_t)i] *= res * (1.f / (1.f + __expf(-res)));
}

// ---------------- host launch ----------------
extern "C" void kernel_launch(void* const* d_in, const int* in_sizes, int n_in,
                              void* d_out, int out_size, void* d_ws, size_t ws_size,
                              hipStream_t stream) {
  const float* x        = (const float*)d_in[0];
  const float* norm_w   = (const float*)d_in[1];
  const float* in_proj  = (const float*)d_in[2];   // [2*D_INNER, HIDDEN]
  const float* conv_w   = (const float*)d_in[3];   // [D_INNER, 1, D_CONV]
  const float* conv_b   = (const float*)d_in[4];
  const float* x_proj   = (const float*)d_in[5];   // [XDBL, D_INNER]
  const float* dt_proj  = (const float*)d_in[6];   // [D_INNER, DT_RANK]
  const float* dt_b     = (const float*)d_in[7];
  const float* A_log    = (const float*)d_in[8];
  const float* Dvec     = (const float*)d_in[9];
  const float* out_proj = (const float*)d_in[10];  // [HIDDEN, D_INNER]
  float* out = (float*)d_out;

  float* wsf   = (float*)d_ws;
  float* xn    = wsf;                                   // ROWS*HIDDEN
  float* yb    = wsf;                                   // ROWS*D_INNER (alias; xn dead)
  float* xr    = wsf + (size_t)ROWS * D_INNER;          // ROWS*2*D_INNER
  float* xc    = xr  + (size_t)ROWS * 2 * D_INNER;      // ROWS*D_INNER
  float* xdbl  = xc  + (size_t)ROWS * D_INNER;          // ROWS*XDBL
  float* delta = xdbl + (size_t)ROWS * XDBL;            // ROWS*D_INNER

  rmsnorm_kernel<<<ROWS, 256, 0, stream>>>(x, norm_w, xn);

  gemm_wmma_f32<<<dim3(2 * D_INNER / 64, ROWS / 128), 256, 0, stream>>>(
      xn, HIDDEN, in_proj, xr, ROWS, 2 * D_INNER, HIDDEN, nullptr, nullptr, 0);

  conv_silu_kernel<<<(ROWS * D_INNER) / 256, 256, 0, stream>>>(xr, conv_w, conv_b, xc);

  gemm_wmma_f32<<<dim3((XDBL + 63) / 64, ROWS / 128), 256, 0, stream>>>(
      xc, D_INNER, x_proj, xdbl, ROWS, XDBL, D_INNER, nullptr, nullptr, 0);

  gemm_wmma_f32<<<dim3(D_INNER / 64, ROWS / 128), 256, 0, stream>>>(
      xdbl, XDBL, dt_proj, delta, ROWS, D_INNER, DT_RANK, dt_b, nullptr, 1);

  selective_scan_kernel<<<(BATCH * D_INNER * STATE) / 256, 256, 0, stream>>>(
      delta, xc, xdbl, A_log, Dvec, yb);

  gate_kernel<<<(ROWS * D_INNER) / 256, 256, 0, stream>>>(xr, yb);

  gemm_wmma_f32<<<dim3(HIDDEN / 64, ROWS / 128), 256, 0, stream>>>(
      yb, D_INNER, out_proj, out, ROWS, HIDDEN, D_INNER, nullptr, x, 2);
}